// Net2_15006615734283
// MI455X (gfx1250) — compile-verified
//
#include <hip/hip_runtime.h>
#include <math.h>

typedef __attribute__((ext_vector_type(2))) float v2f;
typedef __attribute__((ext_vector_type(8))) float v8f;
typedef __attribute__((ext_vector_type(4))) unsigned int v4u;
typedef __attribute__((ext_vector_type(8))) int v8i;
typedef __attribute__((ext_vector_type(4))) int v4i;

#define NB        256                     // batch
#define H1C       10
#define H1HW      62                      // pooled spatial
#define H1IMG     (H1C * H1HW * H1HW)     // 38440 real floats per image
#define H1STRIDE  42304                   // 38440 + 3864 zero pad (>= 38441+3843+1)
#define F_C       20
#define F_N       (58 * 58)               // 3364 pixels per image
#define KTOT      250                     // 10*5*5
#define KPAD      252
#define W2_ELEMS  5000                    // 20*250

// ---------------------------------------------------------------------------
// Kernel 1: conv1 (1->10, 5x5, VALID) + 2x2 maxpool + ReLU, plus zero-fill of
// the per-image K-padding region used by the WMMA implicit GEMM.
// h1 layout: [B][H1STRIDE] where first H1IMG floats are [10][62][62].
// ---------------------------------------------------------------------------
__global__ __launch_bounds__(256) void conv1_pool_relu(
    const float* __restrict__ x, const float* __restrict__ w1,
    const float* __restrict__ b1, float* __restrict__ h1) {
  int idx = blockIdx.x * 256 + threadIdx.x;
  if (idx >= NB * H1STRIDE) return;
  int b   = idx / H1STRIDE;
  int pos = idx - b * H1STRIDE;
  if (pos >= H1IMG) {          // zero pad region (feeds k=250/251 WMMA taps)
    h1[idx] = 0.0f;
    return;
  }
  int px = pos % H1HW;
  int py = (pos / H1HW) % H1HW;
  int o  = pos / (H1HW * H1HW);

  const float* xb = x + (size_t)b * 128 * 128;
  const float* wo = w1 + o * 25;
  float wreg[25];
#pragma unroll
  for (int i = 0; i < 25; ++i) wreg[i] = wo[i];
  float bv = b1[o];

  float best = -INFINITY;
#pragma unroll
  for (int dy = 0; dy < 2; ++dy) {
#pragma unroll
    for (int dx = 0; dx < 2; ++dx) {
      int cy = 2 * py + dy, cx = 2 * px + dx;
      float s = bv;
#pragma unroll
      for (int r = 0; r < 5; ++r)
#pragma unroll
        for (int c = 0; c < 5; ++c)
          s = fmaf(xb[(cy + r) * 128 + (cx + c)], wreg[r * 5 + c], s);
      best = fmaxf(best, s);
    }
  }
  h1[idx] = fmaxf(best, 0.0f);
}

// ---------------------------------------------------------------------------
// Kernel 2: conv2 (10->20, 5x5) as implicit GEMM with V_WMMA_F32_16X16X4_F32.
//   per image:  D[M=3364, N=20] = A[M, K=250] * B[K, N]
//   A[m][k] = h1[b][k/25][m/58 + (k%25)/5][m%58 + k%5]   (k>=250 -> zero pad)
//   B[k][n] = w2[n][k], staged raw into LDS by the Tensor Data Mover.
// Wave tile: 64(M) x 32(N, 20 valid); WG = 8 waves = 512 pixels; 7 chunks/img.
// ---------------------------------------------------------------------------
__global__ __launch_bounds__(256) void conv2_wmma(
    const float* __restrict__ h1, const float* __restrict__ w2,
    const float* __restrict__ b2, float* __restrict__ f) {
  __shared__ float sW[8064];     // [0,5000) = w2 raw via TDM; tail zeroed
  __shared__ int   sKoff[KPAD];  // k -> input offset within an image

  int tid = threadIdx.x;
  int wv  = tid >> 5;

  // zero the LDS tail so padded-k reads can never inject Inf/NaN
  for (int i = W2_ELEMS + tid; i < 8064; i += 256) sW[i] = 0.0f;
  for (int k = tid; k < KPAD; k += 256) {
    int c = k / 25, rem = k % 25;
    int r = rem / 5, s = rem % 5;
    // k >= 250 points into the per-image zero pad after the real data
    sKoff[k] = (k < KTOT) ? (c * (H1HW * H1HW) + r * H1HW + s)
                          : (H1IMG + (k - KTOT));
  }

  // TDM: DMA the 20x250 f32 weight block (5000 elems) global -> LDS (wave 0)
  if (wv == 0) {
    unsigned lds_off = (unsigned)(unsigned long long)(const void*)sW;
    unsigned long long ga = (unsigned long long)(const void*)w2;
    v4u g0;
    g0.x = 1u;                                            // count=1 (valid D#)
    g0.y = lds_off;                                       // lds_addr
    g0.z = (unsigned)(ga & 0xFFFFFFFFull);                // global_addr[31:0]
    g0.w = ((unsigned)(ga >> 32) & 0x01FFFFFFu) | (2u << 30); // addr[56:32]|type=2
    v8i g1;
    g1[0] = (int)(2u << 16);                // data_size=4B, wg_mask=0
    g1[1] = (int)((unsigned)W2_ELEMS << 16);// tensor_dim0[15:0] @bit48
    g1[2] = (int)(1u << 16);                // tensor_dim0 hi=0 | tensor_dim1=1 @bit80
    g1[3] = (int)((unsigned)W2_ELEMS << 16);// tensor_dim1 hi=0 | tile_dim0 @bit112
    g1[4] = 1;                              // tile_dim1=1, tile_dim2=0
    g1[5] = W2_ELEMS;                       // tensor_dim0_stride[31:0]
    g1[6] = 0;
    g1[7] = 0;
    v4i g2 = {0, 0, 0, 0};
    v4i g3 = {0, 0, 0, 0};
    v8i g4 = {0, 0, 0, 0, 0, 0, 0, 0};      // extra SGPR group (6-arg variant)
    __builtin_amdgcn_tensor_load_to_lds(g0, g1, g2, g3, g4, 0);
    __builtin_amdgcn_s_wait_tensorcnt(0);
  }
  __syncthreads();

  int img   = blockIdx.x / 7;
  int chunk = blockIdx.x % 7;
  int lane  = tid & 31;
  int hl    = lane >> 4;      // lane half
  int l16   = lane & 15;
  int m0    = chunk * 512 + wv * 64;

  const float* h1b = h1 + (size_t)img * H1STRIDE;

  int mb[4];
#pragma unroll
  for (int mi = 0; mi < 4; ++mi) {
    int m  = m0 + 16 * mi + l16;
    int mc = m < F_N ? m : (F_N - 1);
    int y  = mc / 58, xx = mc % 58;
    mb[mi] = y * H1HW + xx;
  }

  v8f acc[4][2] = {};

#pragma unroll 3
  for (int k0 = 0; k0 < KPAD; k0 += 4) {
    int kk = k0 + 2 * hl;          // ISA 32-bit A layout: lane halves hold K pairs
    int off0 = sKoff[kk];
    int off1 = sKoff[kk + 1];

    v2f bf[2];
#pragma unroll
    for (int ni = 0; ni < 2; ++ni) {
      int n = 16 * ni + l16;
      bf[ni].x = sW[n * KTOT + kk];       // B[k][n] = w2[n][k]
      bf[ni].y = sW[n * KTOT + kk + 1];
    }

#pragma unroll
    for (int mi = 0; mi < 4; ++mi) {
      v2f a;
      a.x = h1b[off0 + mb[mi]];           // branch-free: pad region is zero
      a.y = h1b[off1 + mb[mi]];
#pragma unroll
      for (int ni = 0; ni < 2; ++ni) {
        acc[mi][ni] = __builtin_amdgcn_wmma_f32_16x16x4_f32(
            false, a, false, bf[ni], (short)0, acc[mi][ni], false, false);
      }
    }
  }

  // store D (+bias): vgpr r holds M = r + 8*half, N = lane%16
  float* fimg = f + (size_t)img * F_C * F_N;
#pragma unroll
  for (int ni = 0; ni < 2; ++ni) {
    int n = 16 * ni + l16;
    if (n < F_C) {
      float bias = b2[n];
#pragma unroll
      for (int mi = 0; mi < 4; ++mi) {
#pragma unroll
        for (int r = 0; r < 8; ++r) {
          int m = m0 + 16 * mi + r + 8 * hl;
          if (m < F_N) fimg[n * F_N + m] = acc[mi][ni][r] + bias;
        }
      }
    }
  }
}

// ---------------------------------------------------------------------------
// Kernel 3: NetVLAD (K=3, D=20) + norms + FC(60->10) + log_softmax.
// One 256-thread block per image; wave shuffle reduce, then LDS atomics.
// ---------------------------------------------------------------------------
__global__ __launch_bounds__(256) void netvlad_head(
    const float* __restrict__ f, const float* __restrict__ aw,
    const float* __restrict__ ab, const float* __restrict__ cents,
    const float* __restrict__ fcw, const float* __restrict__ fcb,
    float* __restrict__ out) {
  __shared__ float sAcc[64];   // [0..59] = S1[k][d], [60..62] = S0[k]
  __shared__ float sAW[64];    // assign_w [3][20]
  int tid  = threadIdx.x;
  int lane = tid & 31;
  int b = blockIdx.x;
  if (tid < 64) {
    sAcc[tid] = 0.0f;
    sAW[tid]  = (tid < 60) ? aw[tid] : 0.0f;
  }
  __syncthreads();

  const float* fb = f + (size_t)b * F_C * F_N;
  float ab0 = ab[0], ab1 = ab[1], ab2 = ab[2];
  float S1[3][20] = {};
  float S0[3] = {};

  for (int n = tid; n < F_N; n += 256) {
    __builtin_prefetch(fb + n + 256, 0, 0);
    float fv[20];
#pragma unroll
    for (int d = 0; d < 20; ++d) fv[d] = fb[d * F_N + n];
    float l0 = ab0, l1 = ab1, l2 = ab2;
#pragma unroll
    for (int d = 0; d < 20; ++d) {
      l0 = fmaf(sAW[d],      fv[d], l0);
      l1 = fmaf(sAW[20 + d], fv[d], l1);
      l2 = fmaf(sAW[40 + d], fv[d], l2);
    }
    float mx = fmaxf(l0, fmaxf(l1, l2));
    float e0 = __expf(l0 - mx), e1 = __expf(l1 - mx), e2 = __expf(l2 - mx);
    float inv = 1.0f / (e0 + e1 + e2);
    float a0 = e0 * inv, a1 = e1 * inv, a2 = e2 * inv;
    S0[0] += a0; S0[1] += a1; S0[2] += a2;
#pragma unroll
    for (int d = 0; d < 20; ++d) {
      S1[0][d] = fmaf(a0, fv[d], S1[0][d]);
      S1[1][d] = fmaf(a1, fv[d], S1[1][d]);
      S1[2][d] = fmaf(a2, fv[d], S1[2][d]);
    }
  }

  // wave32 tree reduction, then one LDS atomic per wave per accumulator
#pragma unroll
  for (int k = 0; k < 3; ++k) {
    float s = S0[k];
#pragma unroll
    for (int off = 16; off > 0; off >>= 1) s += __shfl_xor(s, off, 32);
    if (lane == 0) atomicAdd(&sAcc[60 + k], s);
#pragma unroll
    for (int d = 0; d < 20; ++d) {
      float t = S1[k][d];
#pragma unroll
      for (int off = 16; off > 0; off >>= 1) t += __shfl_xor(t, off, 32);
      if (lane == 0) atomicAdd(&sAcc[k * 20 + d], t);
    }
  }
  __syncthreads();

  if (tid == 0) {
    float v[60];
#pragma unroll
    for (int k = 0; k < 3; ++k) {
      float s0 = sAcc[60 + k];
      float tmp[20];
      float nrm = 0.0f;
#pragma unroll
      for (int d = 0; d < 20; ++d) {
        float t = sAcc[k * 20 + d] - s0 * cents[k * 20 + d];
        tmp[d] = t;
        nrm = fmaf(t, t, nrm);
      }
      nrm = fmaxf(sqrtf(nrm), 1e-12f);
#pragma unroll
      for (int d = 0; d < 20; ++d) v[k * 20 + d] = tmp[d] / nrm;
    }
    float g = 0.0f;
#pragma unroll
    for (int i = 0; i < 60; ++i) g = fmaf(v[i], v[i], g);
    g = fmaxf(sqrtf(g), 1e-12f);
    float inv_g = 1.0f / g;
#pragma unroll
    for (int i = 0; i < 60; ++i) v[i] = fmaxf(v[i] * inv_g, 0.0f);  // relu

    float o[10];
    float mx = -INFINITY;
#pragma unroll
    for (int j = 0; j < 10; ++j) {
      float s = fcb[j];
#pragma unroll
      for (int i = 0; i < 60; ++i) s = fmaf(v[i], fcw[j * 60 + i], s);
      o[j] = s;
      mx = fmaxf(mx, s);
    }
    float se = 0.0f;
#pragma unroll
    for (int j = 0; j < 10; ++j) se += __expf(o[j] - mx);
    float lse = mx + __logf(se);
#pragma unroll
    for (int j = 0; j < 10; ++j) out[b * 10 + j] = o[j] - lse;
  }
}

// ---------------------------------------------------------------------------
extern "C" void kernel_launch(void* const* d_in, const int* in_sizes, int n_in,
                              void* d_out, int out_size, void* d_ws, size_t ws_size,
                              hipStream_t stream) {
  (void)in_sizes; (void)n_in; (void)out_size; (void)ws_size;
  const float* x  = (const float*)d_in[0];
  const float* w1 = (const float*)d_in[1];
  const float* b1 = (const float*)d_in[2];
  const float* w2 = (const float*)d_in[3];
  const float* b2 = (const float*)d_in[4];
  const float* aw = (const float*)d_in[5];
  const float* ab = (const float*)d_in[6];
  const float* ct = (const float*)d_in[7];
  const float* fw = (const float*)d_in[8];
  const float* fb = (const float*)d_in[9];
  float* out = (float*)d_out;

  float* h1 = (float*)d_ws;                    // 256*42304 floats (incl pad)
  float* f  = h1 + (size_t)NB * H1STRIDE;      // 256*20*3364 floats

  int n_h1 = NB * H1STRIDE;
  conv1_pool_relu<<<(n_h1 + 255) / 256, 256, 0, stream>>>(x, w1, b1, h1);
  conv2_wmma<<<NB * 7, 256, 0, stream>>>(h1, w2, b2, f);
  netvlad_head<<<NB, 256, 0, stream>>>(f, aw, ab, ct, fw, fb, out);
}